// MoESystem_15659450761990
// MI455X (gfx1250) — compile-verified
//
#include <hip/hip_runtime.h>
#include <hip/hip_bf16.h>
#include <math.h>

typedef __attribute__((ext_vector_type(2))) float v2f;
typedef __attribute__((ext_vector_type(8))) float v8f;

#define NUM_EXPERTS 16
#define D_STATE     32
#define INPUT_DIM   512
#define COPIES      4
#define BATCH       64
#define HW          1024
#define OUT_ELEMS   (COPIES * INPUT_DIM * D_STATE)   // 65536

// workspace layout (float offsets)
#define WS_X    0        // [64,512] pooled activations
#define WS_LR   32768    // [64,16] route logits (pre-softmax)
#define WS_LN   33792    // [64,16] noise logits
#define WS_COEF 34816    // [16] per-expert combine coefficients
#define WS_Z    34832    // [1] z-loss accumulator

// ---------------------------------------------------------------------------
// Kernel 1: spatial mean pool. One wave32 per (b,c) row of 1024 contiguous
// floats. float4 loads -> global_load_b128; wave32 shfl_xor reduction.
// Bandwidth-dominant pass: 134 MB read.
// ---------------------------------------------------------------------------
__global__ void pool_kernel(const float* __restrict__ in, float* __restrict__ x) {
    int row  = blockIdx.x * 8 + (threadIdx.x >> 5);   // 8 waves/block, 32768 rows
    int lane = threadIdx.x & 31;
    const float4* p = (const float4*)(in + (size_t)row * HW);
    float s = 0.0f;
#pragma unroll
    for (int i = 0; i < 8; ++i) {
        float4 v = p[i * 32 + lane];
        s += v.x + v.y + v.z + v.w;
    }
#pragma unroll
    for (int d = 16; d > 0; d >>= 1) s += __shfl_xor(s, d, 32);
    if (lane == 0) x[row] = s * (1.0f / (float)HW);
}

// ---------------------------------------------------------------------------
// Kernel 2: router GEMMs via V_WMMA_F32_16X16X4_F32 (full-precision f32 WMMA).
// M=64 (4 waves x 16 rows), N=16, K=512 in steps of 4.
// A 16x4 layout: lanes 0-15 row M=lane, VGPR0=K0,VGPR1=K1; lanes 16-31 K2/K3.
// B 4x16 layout: VGPR r holds row K=r (low lanes) / K=r+2 (high lanes), N=lane&15.
// C/D: VGPR r -> M = r (low lanes) / r+8 (high lanes), N = lane&15.
// ---------------------------------------------------------------------------
__global__ void router_gemm_kernel(const float* __restrict__ X,   // [64,512]
                                   const float* __restrict__ Wr,  // [512,16]
                                   const float* __restrict__ br,  // [16]
                                   const float* __restrict__ Wn,  // [512,16]
                                   const float* __restrict__ bn,  // [16]
                                   float* __restrict__ Lr,        // [64,16]
                                   float* __restrict__ Ln) {      // [64,16]
    int wave = threadIdx.x >> 5;         // 0..3 -> M tile
    int lane = threadIdx.x & 31;
    int half = lane >> 4;                // 0 | 1
    int l    = lane & 15;

    v8f cr = {};
    v8f cn = {};
    const float* xrow = X + (size_t)(wave * 16 + l) * INPUT_DIM;

    for (int k0 = 0; k0 < INPUT_DIM; k0 += 4) {
        int ka = k0 + half * 2;
        v2f a;  a.x = xrow[ka];           a.y = xrow[ka + 1];
        v2f b0; b0.x = Wr[ka * 16 + l];   b0.y = Wr[(ka + 1) * 16 + l];
        v2f b1; b1.x = Wn[ka * 16 + l];   b1.y = Wn[(ka + 1) * 16 + l];
        cr = __builtin_amdgcn_wmma_f32_16x16x4_f32(false, a, false, b0,
                                                   (short)0, cr, false, false);
        cn = __builtin_amdgcn_wmma_f32_16x16x4_f32(false, a, false, b1,
                                                   (short)0, cn, false, false);
    }

#pragma unroll
    for (int r = 0; r < 8; ++r) {
        int row = wave * 16 + r + half * 8;
        Lr[row * 16 + l] = cr[r] + br[l];
        Ln[row * 16 + l] = cn[r] + bn[l];
    }
}

// ---------------------------------------------------------------------------
// Threefry2x32 (JAX default PRNG) + erfinv for jax.random.normal(key(42)).
// ---------------------------------------------------------------------------
__device__ __forceinline__ unsigned rotl32(unsigned x, int d) {
    return (x << d) | (x >> (32 - d));
}

__device__ void threefry2x32(unsigned k0, unsigned k1, unsigned x0, unsigned x1,
                             unsigned& o0, unsigned& o1) {
    const int R[8] = {13, 15, 26, 6, 17, 29, 16, 24};
    unsigned ks[3] = {k0, k1, k0 ^ k1 ^ 0x1BD11BDAu};
    x0 += ks[0]; x1 += ks[1];
#pragma unroll
    for (int g = 1; g <= 5; ++g) {
        const int* r = (g & 1) ? R : (R + 4);
#pragma unroll
        for (int i = 0; i < 4; ++i) { x0 += x1; x1 = rotl32(x1, r[i]); x1 ^= x0; }
        x0 += ks[g % 3];
        x1 += ks[(g + 1) % 3] + (unsigned)g;
    }
    o0 = x0; o1 = x1;
}

__device__ float erfinv_f(float x) {   // Giles 2010 single-precision approx
    float w = -logf((1.0f - x) * (1.0f + x));
    float p;
    if (w < 5.0f) {
        w -= 2.5f;
        p = 2.81022636e-08f;
        p = 3.43273939e-07f + p * w;
        p = -3.5233877e-06f + p * w;
        p = -4.39150654e-06f + p * w;
        p = 0.00021858087f  + p * w;
        p = -0.00125372503f + p * w;
        p = -0.00417768164f + p * w;
        p = 0.246640727f    + p * w;
        p = 1.50140941f     + p * w;
    } else {
        w = sqrtf(w) - 3.0f;
        p = -0.000200214257f;
        p = 0.000100950558f + p * w;
        p = 0.00134934322f  + p * w;
        p = -0.00367342844f + p * w;
        p = 0.00573950773f  + p * w;
        p = -0.0076224613f  + p * w;
        p = 0.00943887047f  + p * w;
        p = 1.00167406f     + p * w;
        p = 2.83297682f     + p * w;
    }
    return p * x;
}

// ---------------------------------------------------------------------------
// Kernel 3: router post-processing. One thread per batch row (64 threads).
// softmax(logits); noise = softmax(eps * softplus(noise_logits));
// top-2 of (softmax + noise); softmax over top-2 -> per-expert coef atomics;
// z-loss = sum(logsumexp^2)/B.
// ---------------------------------------------------------------------------
__global__ void router_post_kernel(const float* __restrict__ Lr,
                                   const float* __restrict__ Ln,
                                   float* __restrict__ coef,
                                   float* __restrict__ zacc) {
    int b = threadIdx.x;
    if (b >= BATCH) return;

    float lg[NUM_EXPERTS], nl[NUM_EXPERTS], t[NUM_EXPERTS], noisy[NUM_EXPERTS];
#pragma unroll
    for (int e = 0; e < NUM_EXPERTS; ++e) {
        lg[e] = Lr[b * NUM_EXPERTS + e];
        nl[e] = Ln[b * NUM_EXPERTS + e];
    }
    // softmax(logits)
    float m = -1e30f;
#pragma unroll
    for (int e = 0; e < NUM_EXPERTS; ++e) m = fmaxf(m, lg[e]);
    float s = 0.0f;
#pragma unroll
    for (int e = 0; e < NUM_EXPERTS; ++e) { lg[e] = expf(lg[e] - m); s += lg[e]; }
#pragma unroll
    for (int e = 0; e < NUM_EXPERTS; ++e) lg[e] /= s;

    // eps[b,e] via threefry2x32(key=(0,42)), JAX half-split counter convention
#pragma unroll
    for (int e = 0; e < NUM_EXPERTS; ++e) {
        int i = b * NUM_EXPERTS + e;          // 0..1023
        unsigned x0 = (i < 512) ? (unsigned)i : (unsigned)(i - 512);
        unsigned x1 = (i < 512) ? (unsigned)(i + 512) : (unsigned)i;
        unsigned r0, r1;
        threefry2x32(0u, 42u, x0, x1, r0, r1);
        unsigned bits = (i < 512) ? r0 : r1;
        float u01 = __uint_as_float((bits >> 9) | 0x3f800000u) - 1.0f;  // [0,1)
        float un  = fminf(fmaxf(2.0f * u01 - 1.0f, -0.9999999f), 0.9999999f);
        float eps = 1.41421356237f * erfinv_f(un);
        float sp  = (nl[e] > 20.0f) ? nl[e] : log1pf(expf(nl[e]));       // softplus
        t[e] = eps * sp;
    }
    // softmax(noise term)
    m = -1e30f;
#pragma unroll
    for (int e = 0; e < NUM_EXPERTS; ++e) m = fmaxf(m, t[e]);
    s = 0.0f;
#pragma unroll
    for (int e = 0; e < NUM_EXPERTS; ++e) { t[e] = expf(t[e] - m); s += t[e]; }
#pragma unroll
    for (int e = 0; e < NUM_EXPERTS; ++e) noisy[e] = lg[e] + t[e] / s;

    // top-2 (first occurrence wins ties, matching lax.top_k)
    int i0 = 0; float v0 = -1e30f;
#pragma unroll
    for (int e = 0; e < NUM_EXPERTS; ++e) if (noisy[e] > v0) { v0 = noisy[e]; i0 = e; }
    int i1 = 0; float v1 = -1e30f;
#pragma unroll
    for (int e = 0; e < NUM_EXPERTS; ++e)
        if (e != i0 && noisy[e] > v1) { v1 = noisy[e]; i1 = e; }

    // softmax over (v0, v1)
    float e1 = expf(v1 - v0);
    float w0 = 1.0f / (1.0f + e1);
    float w1 = e1 / (1.0f + e1);
    atomicAdd(&coef[i0], w0);
    atomicAdd(&coef[i1], w1);

    // z-loss contribution: logsumexp(noisy)^2
    float ss = 0.0f;
#pragma unroll
    for (int e = 0; e < NUM_EXPERTS; ++e) ss += expf(noisy[e] - v0);
    float lse = v0 + logf(ss);
    atomicAdd(zacc, lse * lse);
}

// ---------------------------------------------------------------------------
// Kernel 4: combine = (sum_e coef[e] * A_logs[e]) / B ; append z_loss.
// ---------------------------------------------------------------------------
__global__ void combine_kernel(const float* __restrict__ Alogs,  // [16,2048,32]
                               const float* __restrict__ coef,
                               const float* __restrict__ zacc,
                               float* __restrict__ out) {
    int t = blockIdx.x * blockDim.x + threadIdx.x;   // 0..65535
    float s = 0.0f;
#pragma unroll
    for (int e = 0; e < NUM_EXPERTS; ++e)
        s += coef[e] * Alogs[(size_t)e * OUT_ELEMS + t];
    out[t] = s * (1.0f / (float)BATCH);
    if (t == 0) out[OUT_ELEMS] = zacc[0] * (1.0f / (float)BATCH);
}

// ---------------------------------------------------------------------------
extern "C" void kernel_launch(void* const* d_in, const int* in_sizes, int n_in,
                              void* d_out, int out_size, void* d_ws, size_t ws_size,
                              hipStream_t stream) {
    const float* inputs = (const float*)d_in[0];   // [64,512,32,32]
    const float* Wr     = (const float*)d_in[1];   // [512,16]
    const float* br     = (const float*)d_in[2];   // [16]
    const float* Wn     = (const float*)d_in[3];   // [512,16]
    const float* bn     = (const float*)d_in[4];   // [16]
    const float* Alogs  = (const float*)d_in[5];   // [16,2048,32]
    float* ws  = (float*)d_ws;
    float* out = (float*)d_out;

    // zero coef[16] + zacc[1] each call (graph-capturable memset node)
    hipMemsetAsync(ws + WS_COEF, 0, (NUM_EXPERTS + 1) * sizeof(float), stream);

    pool_kernel<<<(BATCH * INPUT_DIM) / 8, 256, 0, stream>>>(inputs, ws + WS_X);
    router_gemm_kernel<<<1, 128, 0, stream>>>(ws + WS_X, Wr, br, Wn, bn,
                                              ws + WS_LR, ws + WS_LN);
    router_post_kernel<<<1, 64, 0, stream>>>(ws + WS_LR, ws + WS_LN,
                                             ws + WS_COEF, ws + WS_Z);
    combine_kernel<<<OUT_ELEMS / 256, 256, 0, stream>>>(Alogs, ws + WS_COEF,
                                                        ws + WS_Z, out);
}